// ECCLoss_7026566496476
// MI455X (gfx1250) — compile-verified
//
#include <hip/hip_runtime.h>
#include <hip/hip_bf16.h>
#include <math.h>

#define NUM_CLASS 1000
#define DIM 1024
#define BATCH 16384
#define NTILE 63          // ceil(1000/16)
#define SPAD 1008         // padded row stride of scratch similarity matrix (63*16)
#define EPS 1e-8f

typedef __attribute__((ext_vector_type(2))) float v2f;
typedef __attribute__((ext_vector_type(8))) float v8f;

// ---------------- block reduction helpers (deterministic tree) ----------------
__device__ __forceinline__ float block_reduce_sum(float v, float* red, int tid) {
  red[tid] = v; __syncthreads();
  for (int o = 128; o > 0; o >>= 1) { if (tid < o) red[tid] += red[tid + o]; __syncthreads(); }
  float r = red[0]; __syncthreads();
  return r;
}
__device__ __forceinline__ float block_reduce_max(float v, float* red, int tid) {
  red[tid] = v; __syncthreads();
  for (int o = 128; o > 0; o >>= 1) { if (tid < o) red[tid] = fmaxf(red[tid], red[tid + o]); __syncthreads(); }
  float r = red[0]; __syncthreads();
  return r;
}

// ---------------- K1: per-class mean of feature rows -> feature_table ----------------
__global__ __launch_bounds__(256) void k_feature_table(
    const float* __restrict__ feat, const int* __restrict__ targets,
    const float* __restrict__ init, float* __restrict__ table)
{
  const int c = blockIdx.x;
  const int tid = threadIdx.x;
  __shared__ int tg[4096];
  float a0 = 0.f, a1 = 0.f, a2 = 0.f, a3 = 0.f;
  int cnt = 0;
  for (int base = 0; base < BATCH; base += 4096) {
    for (int i = tid; i < 4096; i += 256) tg[i] = targets[base + i];
    __syncthreads();
    for (int bb = 0; bb < 4096; ++bb) {
      if (tg[bb] == c) {
        const float* row = feat + (size_t)(base + bb) * DIM;
        a0 += row[tid];
        a1 += row[tid + 256];
        a2 += row[tid + 512];
        a3 += row[tid + 768];
        ++cnt;
      }
    }
    __syncthreads();
  }
  const float inv = 1.0f / (float)(cnt > 0 ? cnt : 1);
  float* dst = table + (size_t)c * DIM;
  const float* src = init + (size_t)c * DIM;
  dst[tid      ] = cnt ? a0 * inv : src[tid      ];
  dst[tid + 256] = cnt ? a1 * inv : src[tid + 256];
  dst[tid + 512] = cnt ? a2 * inv : src[tid + 512];
  dst[tid + 768] = cnt ? a3 * inv : src[tid + 768];
}

// ---------------- K2: per-class mean of logit rows -> logit_table ----------------
__global__ __launch_bounds__(256) void k_logit_table(
    const float* __restrict__ lg, const int* __restrict__ targets,
    const float* __restrict__ init, float* __restrict__ table)
{
  const int c = blockIdx.x;
  const int tid = threadIdx.x;
  __shared__ int tg[4096];
  float a0 = 0.f, a1 = 0.f, a2 = 0.f, a3 = 0.f;
  int cnt = 0;
  for (int base = 0; base < BATCH; base += 4096) {
    for (int i = tid; i < 4096; i += 256) tg[i] = targets[base + i];
    __syncthreads();
    for (int bb = 0; bb < 4096; ++bb) {
      if (tg[bb] == c) {
        const float* row = lg + (size_t)(base + bb) * NUM_CLASS;
        a0 += row[tid];
        a1 += row[tid + 256];
        a2 += row[tid + 512];
        if (tid < NUM_CLASS - 768) a3 += row[tid + 768];
        ++cnt;
      }
    }
    __syncthreads();
  }
  const float inv = 1.0f / (float)(cnt > 0 ? cnt : 1);
  float* dst = table + (size_t)c * NUM_CLASS;
  const float* src = init + (size_t)c * NUM_CLASS;
  dst[tid      ] = cnt ? a0 * inv : src[tid      ];
  dst[tid + 256] = cnt ? a1 * inv : src[tid + 256];
  dst[tid + 512] = cnt ? a2 * inv : src[tid + 512];
  if (tid < NUM_CLASS - 768) dst[tid + 768] = cnt ? a3 * inv : src[tid + 768];
}

// ---------------- K3: row L2 norms (and reciprocals) of feature_table ----------------
__global__ __launch_bounds__(256) void k_row_norm(const float* __restrict__ table,
                                                  float* __restrict__ norms,
                                                  float* __restrict__ rnorms)
{
  const int c = blockIdx.x, tid = threadIdx.x;
  float s = 0.f;
  const float* row = table + (size_t)c * DIM;
  for (int d = tid; d < DIM; d += 256) { float v = row[d]; s += v * v; }
  __shared__ float red[256];
  s = block_reduce_sum(s, red, tid);
  if (tid == 0) {
    const float n = sqrtf(s);
    norms[c] = n;
    rnorms[c] = 1.0f / n;     // 1000 divisions total (instead of 1M in the GEMM epilogue)
  }
}

// ---------------- K4: S = normalize(F @ F^T) via V_WMMA_F32_16X16X4_F32 ----------------
// One wave computes TWO 16x16 output tiles (shared A, two B tiles -> two
// independent accumulation chains). Rows are CLAMPED (not masked): an OOB A/B
// row only pollutes output rows/cols >= NUM_CLASS, which land in the pad region
// of the SPAD-stride scratch matrix, except cols >= SPAD, which wrap into the next row. This keeps EXEC
// untouched: b64 loads in the loop, unconditional stores in the epilogue.
__global__ __launch_bounds__(32) void k_gemm_wmma(const float* __restrict__ ft,
                                                  const float* __restrict__ rnorms,
                                                  float* __restrict__ S)
{
  const int lane = threadIdx.x;          // 0..31 (wave32)
  const int half = lane >> 4;            // 0: K={0,1}, 1: K={2,3}
  const int l15  = lane & 15;
  const int m0 = blockIdx.y * 16;
  const int n0 = blockIdx.x * 32;

  int rowA  = m0 + l15;        if (rowA  > NUM_CLASS - 1) rowA  = NUM_CLASS - 1;
  int rowB0 = n0 + l15;        if (rowB0 > NUM_CLASS - 1) rowB0 = NUM_CLASS - 1;
  int rowB1 = n0 + 16 + l15;   if (rowB1 > NUM_CLASS - 1) rowB1 = NUM_CLASS - 1;

  const v2f* pa  = (const v2f*)(ft + (size_t)rowA  * DIM) + half;
  const v2f* pb0 = (const v2f*)(ft + (size_t)rowB0 * DIM) + half;
  const v2f* pb1 = (const v2f*)(ft + (size_t)rowB1 * DIM) + half;

  v8f c0 = {}, c1 = {};
#pragma unroll 4
  for (int k = 0; k < DIM / 2; k += 2) {   // each iter covers K-slab of 4
    v2f a  = pa[k];
    v2f b0 = pb0[k];
    v2f b1 = pb1[k];
    c0 = __builtin_amdgcn_wmma_f32_16x16x4_f32(false, a, false, b0, (short)0, c0, false, false);
    c1 = __builtin_amdgcn_wmma_f32_16x16x4_f32(false, a, false, b1, (short)0, c1, false, false);
  }

  // Epilogue: pure multiplies, no guards. gm may reach 1007 and gn 1023; rnorms[] slots
  // are 1024 wide in the workspace and stores past NUM_CLASS hit pad cells; gn >= SPAD wraps to row gm+1.
  const int gn0 = n0 + l15;
  const int gn1 = n0 + 16 + l15;
  const float rn0 = rnorms[gn0];
  const float rn1 = rnorms[gn1];
#pragma unroll
  for (int v = 0; v < 8; ++v) {
    const int gm = m0 + v + 8 * half;
    const float rm = rnorms[gm];
    S[(size_t)gm * SPAD + gn0] = c0[v] * rm * rn0;
    S[(size_t)gm * SPAD + gn1] = c1[v] * rm * rn1;
  }
}

// ---------------- K5a/b: global min/max of S (valid region only) ----------------
__global__ __launch_bounds__(256) void k_minmax_part(const float* __restrict__ S,
                                                     float* __restrict__ pmin,
                                                     float* __restrict__ pmax)
{
  const int tid = threadIdx.x;
  float mn = 3.0e38f, mx = -3.0e38f;
  for (int e = blockIdx.x * 256 + tid; e < NUM_CLASS * NUM_CLASS; e += 256 * 1024) {
    int i = e / NUM_CLASS;
    int j = e - i * NUM_CLASS;
    float v = S[(size_t)i * SPAD + j];
    mn = fminf(mn, v); mx = fmaxf(mx, v);
  }
  __shared__ float rmn[256], rmx[256];
  rmn[tid] = mn; rmx[tid] = mx; __syncthreads();
  for (int o = 128; o > 0; o >>= 1) {
    if (tid < o) { rmn[tid] = fminf(rmn[tid], rmn[tid + o]); rmx[tid] = fmaxf(rmx[tid], rmx[tid + o]); }
    __syncthreads();
  }
  if (tid == 0) { pmin[blockIdx.x] = rmn[0]; pmax[blockIdx.x] = rmx[0]; }
}

__global__ __launch_bounds__(256) void k_minmax_final(const float* __restrict__ pmin,
                                                      const float* __restrict__ pmax,
                                                      float* __restrict__ scal)
{
  const int tid = threadIdx.x;
  float mn = 3.0e38f, mx = -3.0e38f;
  for (int i = tid; i < 1024; i += 256) { mn = fminf(mn, pmin[i]); mx = fmaxf(mx, pmax[i]); }
  __shared__ float rmn[256], rmx[256];
  rmn[tid] = mn; rmx[tid] = mx; __syncthreads();
  for (int o = 128; o > 0; o >>= 1) {
    if (tid < o) { rmn[tid] = fminf(rmn[tid], rmn[tid + o]); rmx[tid] = fmaxf(rmx[tid], rmx[tid + o]); }
    __syncthreads();
  }
  if (tid == 0) { scal[0] = rmn[0]; scal[1] = rmx[0]; }
}

// ---------------- K6: per-row max/argmax of scaled S with zeroed diagonal ----------------
__global__ __launch_bounds__(256) void k_rowmax(const float* __restrict__ S,
                                                const float* __restrict__ scal,
                                                float* __restrict__ simval,
                                                int* __restrict__ simidx)
{
  const int i = blockIdx.x, tid = threadIdx.x;
  const float cmin = scal[0];
  const float inv = 1.0f / (scal[1] - cmin);
  float bv = -3.0e38f; int bj = 0x7fffffff;
  for (int j = tid; j < NUM_CLASS; j += 256) {
    float v = (j == i) ? 0.0f : (S[(size_t)i * SPAD + j] - cmin) * inv;
    if (v > bv) { bv = v; bj = j; }               // first occurrence per-thread (ascending j)
  }
  __shared__ float sv[256]; __shared__ int sj[256];
  sv[tid] = bv; sj[tid] = bj; __syncthreads();
  for (int o = 128; o > 0; o >>= 1) {
    if (tid < o) {
      if (sv[tid + o] > sv[tid] || (sv[tid + o] == sv[tid] && sj[tid + o] < sj[tid])) {
        sv[tid] = sv[tid + o]; sj[tid] = sj[tid + o];
      }
    }
    __syncthreads();
  }
  if (tid == 0) { simval[i] = sv[0]; simidx[i] = sj[0]; }
}

// ---------------- K7: per-sample feature-center + intra losses ----------------
__global__ __launch_bounds__(256) void k_feat_loss(
    const float* __restrict__ feat, const int* __restrict__ targets,
    const float* __restrict__ ftab, const float* __restrict__ norms,
    const float* __restrict__ simval, const int* __restrict__ simidx,
    float* __restrict__ fc, float* __restrict__ fi)
{
  const int b = blockIdx.x, tid = threadIdx.x;
  const int t = targets[b];
  const int s = simidx[t];
  const float sv = simval[t];
  const float* fb = feat + (size_t)b * DIM;
  const float* rt = ftab + (size_t)t * DIM;
  const float* rs = ftab + (size_t)s * DIM;
  float d1 = 0.f, d2 = 0.f, nf2 = 0.f;
  for (int d = tid; d < DIM; d += 256) {
    float x = fb[d];
    d1 += rt[d] * x;
    d2 += rs[d] * x;
    nf2 += x * x;
  }
  __shared__ float red[256];
  d1  = block_reduce_sum(d1, red, tid);
  d2  = block_reduce_sum(d2, red, tid);
  nf2 = block_reduce_sum(nf2, red, tid);
  if (tid == 0) {
    const float nF = fmaxf(sqrtf(nf2), EPS);
    const float nT = fmaxf(norms[t], EPS);
    const float nS = fmaxf(norms[s], EPS);
    fc[b] = 1.0f - d1 / (nT * nF);
    fi[b] = (d2 / (nS * nF)) * sv;
  }
}

// ---------------- K8: per-sample KL(softmax(logit_table[t]) || softmax(logits)) ----------------
__global__ __launch_bounds__(256) void k_kl(
    const float* __restrict__ logits, const int* __restrict__ targets,
    const float* __restrict__ ltab, float* __restrict__ kl)
{
  const int b = blockIdx.x, tid = threadIdx.x;
  const int t = targets[b];
  const float* q = logits + (size_t)b * NUM_CLASS;
  const float* p = ltab + (size_t)t * NUM_CLASS;
  __shared__ float red[256];
  float mp = -3.0e38f, mq = -3.0e38f;
  for (int j = tid; j < NUM_CLASS; j += 256) { mp = fmaxf(mp, p[j]); mq = fmaxf(mq, q[j]); }
  mp = block_reduce_max(mp, red, tid);
  mq = block_reduce_max(mq, red, tid);
  float sp = 0.f, sq = 0.f;
  for (int j = tid; j < NUM_CLASS; j += 256) { sp += expf(p[j] - mp); sq += expf(q[j] - mq); }
  sp = block_reduce_sum(sp, red, tid);
  sq = block_reduce_sum(sq, red, tid);
  const float lsp = logf(sp), lsq = logf(sq);
  float acc = 0.f;
  for (int j = tid; j < NUM_CLASS; j += 256) {
    float lp = p[j] - mp - lsp;
    float lq = q[j] - mq - lsq;
    acc += expf(lp) * (lp - lq);
  }
  acc = block_reduce_sum(acc, red, tid);
  if (tid == 0) kl[b] = acc;
}

// ---------------- K9: final deterministic sums -> out[0], out[1] ----------------
__global__ __launch_bounds__(256) void k_final(const float* __restrict__ fc,
                                               const float* __restrict__ fi,
                                               const float* __restrict__ kl,
                                               float* __restrict__ out)
{
  const int tid = threadIdx.x;
  float s1 = 0.f, s2 = 0.f;
  for (int b = tid; b < BATCH; b += 256) { s1 += fc[b] + fi[b]; s2 += kl[b]; }
  __shared__ float red[256];
  s1 = block_reduce_sum(s1, red, tid);
  s2 = block_reduce_sum(s2, red, tid);
  if (tid == 0) { out[0] = s1; out[1] = s2; }
}

extern "C" void kernel_launch(void* const* d_in, const int* in_sizes, int n_in,
                              void* d_out, int out_size, void* d_ws, size_t ws_size,
                              hipStream_t stream) {
  const float* feature = (const float*)d_in[0];
  const float* logits  = (const float*)d_in[1];
  const int*   targets = (const int*)d_in[2];
  const float* ft_init = (const float*)d_in[3];
  const float* lt_init = (const float*)d_in[4];

  float* out = (float*)d_out;
  float* ft  = out + 2;                                  // feature_table [1000 x 1024]
  float* lt  = out + 2 + (size_t)NUM_CLASS * DIM;        // logit_table   [1000 x 1000]

  float* ws        = (float*)d_ws;
  float* ws_norm   = ws + 1024;                          // [1000] (1024 slots)
  float* ws_simval = ws + 2048;                          // [1000]
  int*   ws_simidx = (int*)(ws + 3072);                  // [1000]
  float* ws_minp   = ws + 4096;                          // [1024]
  float* ws_maxp   = ws + 5120;                          // [1024]
  float* ws_scal   = ws + 6144;                          // [2]
  float* ws_rnorm  = ws + 7168;                          // [1000] (1024 slots; pad reads OK)
  float* ws_fc     = ws + 8192;                          // [BATCH]
  float* ws_fi     = ws + 8192 + BATCH;                  // [BATCH]
  float* ws_kl     = ws + 8192 + 2 * BATCH;              // [BATCH]
  float* ws_S      = ws + 8192 + 3 * BATCH;              // [SPAD x SPAD] padded

  k_feature_table<<<NUM_CLASS, 256, 0, stream>>>(feature, targets, ft_init, ft);
  k_logit_table  <<<NUM_CLASS, 256, 0, stream>>>(logits, targets, lt_init, lt);
  k_row_norm     <<<NUM_CLASS, 256, 0, stream>>>(ft, ws_norm, ws_rnorm);
  k_gemm_wmma    <<<dim3(32, NTILE), 32, 0, stream>>>(ft, ws_rnorm, ws_S);
  k_minmax_part  <<<1024, 256, 0, stream>>>(ws_S, ws_minp, ws_maxp);
  k_minmax_final <<<1, 256, 0, stream>>>(ws_minp, ws_maxp, ws_scal);
  k_rowmax       <<<NUM_CLASS, 256, 0, stream>>>(ws_S, ws_scal, ws_simval, ws_simidx);
  k_feat_loss    <<<BATCH, 256, 0, stream>>>(feature, targets, ft, ws_norm, ws_simval, ws_simidx, ws_fc, ws_fi);
  k_kl           <<<BATCH, 256, 0, stream>>>(logits, targets, lt, ws_kl);
  k_final        <<<1, 256, 0, stream>>>(ws_fc, ws_fi, ws_kl, out);
}